// TXCDRDynamics_16612933501392
// MI455X (gfx1250) — compile-verified
//
#include <hip/hip_runtime.h>
#include <math.h>

typedef float v2f __attribute__((ext_vector_type(2)));
typedef float v8f __attribute__((ext_vector_type(8)));

#define D_INP   768
#define D_SAE   12288
#define T_STEPS 64
#define K_TOP   64
#define BATCH   32
#define M_TOTAL (BATCH * T_STEPS)   // 2048

// ---------------------------------------------------------------------------
// Encode: pre[M=2048, N=12288] = x[M,768] @ W_enc[768,N] + b_enc, full fp32
// via V_WMMA_F32_16X16X4_F32. One wave computes a 32(M) x 64(N) strip:
// each B operand (4x16, 256B) feeds TWO WMMAs (M-lo / M-hi), halving the
// dominant B-side L2 traffic vs a 16x64 tile.
// ---------------------------------------------------------------------------
__global__ __launch_bounds__(128)
void encode_wmma(const float* __restrict__ x, const float* __restrict__ W,
                 const float* __restrict__ b_enc, float* __restrict__ pre) {
  const int lane  = threadIdx.x & 31;
  const int wave  = threadIdx.x >> 5;
  const int mtile = blockIdx.y * 32;                 // 2048/32 = 64
  const int n0    = blockIdx.x * 256 + wave * 64;    // 12288/256 = 48 blocks x 4 waves
  const int mrow  = lane & 15;                       // A: M index within sub-tile
  const int kb    = (lane >> 4) * 2;                 // A/B: K sub-offset (lanes 16-31 -> K+2)
  const int ncol  = lane & 15;                       // B/C: N index

  v8f acc[8] = {};                                   // [0..3]: M-lo tiles, [4..7]: M-hi
  const float* arow0 = x + (size_t)(mtile + mrow) * D_INP;        // M in [0,16)
  const float* arow1 = arow0 + (size_t)16 * D_INP;                // M in [16,32)

  for (int k = 0; k < D_INP; k += 4) {
    v2f a0 = *(const v2f*)(arow0 + k + kb);          // A[m,    k+kb..k+kb+1]
    v2f a1 = *(const v2f*)(arow1 + k + kb);          // A[m+16, k+kb..k+kb+1]
    const float* brow0 = W + (size_t)(k + kb) * D_SAE;
    const float* brow1 = brow0 + D_SAE;
#pragma unroll
    for (int t = 0; t < 4; ++t) {
      int c = n0 + t * 16 + ncol;
      v2f b;
      b.x = brow0[c];                                // B[k+kb,   n]
      b.y = brow1[c];                                // B[k+kb+1, n]
      acc[t] = __builtin_amdgcn_wmma_f32_16x16x4_f32(
          false, a0, false, b, (short)0, acc[t], false, false);
      acc[4 + t] = __builtin_amdgcn_wmma_f32_16x16x4_f32(
          false, a1, false, b, (short)0, acc[4 + t], false, false);
    }
  }

#pragma unroll
  for (int t = 0; t < 4; ++t) {
    int c = n0 + t * 16 + ncol;
    float bias = b_enc[c];
#pragma unroll
    for (int j = 0; j < 8; ++j) {
      int r = mtile + j + ((lane >> 4) * 8);         // C vgpr j: M = j (+8 for hi lanes)
      pre[(size_t)r * D_SAE + c]        = acc[t][j]     + bias;
      pre[(size_t)(r + 16) * D_SAE + c] = acc[4 + t][j] + bias;
    }
  }
}

// ---------------------------------------------------------------------------
// Sequential gated top-K scan. One block per batch row, serial over T.
// Byte-radix select (4 x 256-bin histogram) for the K-th largest, then
// deterministic chunked prefix-scan selection (index order, no atomics).
// ---------------------------------------------------------------------------
__device__ __forceinline__ unsigned f2key(float f) {
  unsigned u = __float_as_uint(f);
  return (u & 0x80000000u) ? ~u : (u | 0x80000000u); // order-preserving
}

__global__ __launch_bounds__(512)
void scan_topk(const float* __restrict__ pre, const float* __restrict__ gate_raw,
               int* __restrict__ z_idx, float* __restrict__ z_val,
               float* __restrict__ zlast) {
  __shared__ float    cand[D_SAE];       // 48 KB candidate vector
  __shared__ unsigned hist[256];
  __shared__ unsigned scanA[512];
  __shared__ unsigned scanB[512];
  __shared__ int      cur_idx[K_TOP];
  __shared__ float    cur_val[K_TOP];
  __shared__ int      prev_idx[K_TOP];
  __shared__ float    prev_val[K_TOP];
  __shared__ unsigned s_prefix, s_remaining;

  const int tid = threadIdx.x;
  const int b   = blockIdx.x;
  const int CH  = D_SAE / 512;           // 24 contiguous elements per thread

  for (int t = 0; t < T_STEPS; ++t) {
    // 1) candidate = pre[b,t,:]
    const float* row = pre + (size_t)(b * T_STEPS + t) * D_SAE;
    for (int j = tid; j < D_SAE; j += 512) cand[j] = row[j];
    __syncthreads();

    // 2) + gate * z_prev (sparse, 64 distinct positions)
    if (t > 0 && tid < K_TOP) {
      int s = prev_idx[tid];
      float g = 1.0f / (1.0f + __expf(-gate_raw[s]));
      cand[s] += g * prev_val[tid];
    }
    if (tid == 0) { s_prefix = 0u; s_remaining = K_TOP; }
    __syncthreads();

    // 3) radix select K-th largest key, byte by byte (MSB first)
    for (int p = 3; p >= 0; --p) {
      const int shift = 8 * p;
      const unsigned maskAbove = (p == 3) ? 0u : (0xFFFFFFFFu << (shift + 8));
      if (tid < 256) hist[tid] = 0u;
      __syncthreads();
      unsigned pref = s_prefix;
      for (int e = 0; e < CH; ++e) {
        unsigned key = f2key(cand[tid * CH + e]);
        if ((key & maskAbove) == (pref & maskAbove))
          atomicAdd(&hist[(key >> shift) & 255u], 1u);
      }
      __syncthreads();
      if (tid == 0) {
        unsigned rem = s_remaining, acc = 0u;
        int chosen = 0;
        for (int bin = 255; bin >= 0; --bin) {
          unsigned c = hist[bin];
          if (acc + c >= rem) { chosen = bin; break; }
          acc += c;
        }
        s_prefix    = pref | ((unsigned)chosen << shift);
        s_remaining = rem - acc;
      }
      __syncthreads();
    }
    const unsigned thr = s_prefix;

    // 4) per-chunk counts of (> thr) and (== thr)
    unsigned cg = 0, ce = 0;
    for (int e = 0; e < CH; ++e) {
      unsigned key = f2key(cand[tid * CH + e]);
      if (key > thr) ++cg; else if (key == thr) ++ce;
    }

    // inclusive scan of cg (double-buffered Hillis-Steele, 512 entries)
    scanA[tid] = cg; __syncthreads();
    unsigned *src = scanA, *dst = scanB;
    for (int off = 1; off < 512; off <<= 1) {
      unsigned v = src[tid];
      if (tid >= off) v += src[tid - off];
      dst[tid] = v; __syncthreads();
      unsigned* tmp = src; src = dst; dst = tmp;
    }
    const unsigned og = src[tid] - cg;
    const unsigned G  = src[511];        // total strictly-greater (<= 63)
    __syncthreads();

    // inclusive scan of ce
    scanA[tid] = ce; __syncthreads();
    src = scanA; dst = scanB;
    for (int off = 1; off < 512; off <<= 1) {
      unsigned v = src[tid];
      if (tid >= off) v += src[tid - off];
      dst[tid] = v; __syncthreads();
      unsigned* tmp = src; src = dst; dst = tmp;
    }
    const unsigned oe   = src[tid] - ce;
    const unsigned need = K_TOP - G;     // ties to take, in index order
    __syncthreads();

    // 5) deterministic write of the K selected (idx, relu(val))
    {
      unsigned g = og, e = oe;
      for (int el = 0; el < CH; ++el) {
        int j = tid * CH + el;
        float v = cand[j];
        unsigned key = f2key(v);
        if (key > thr) {
          cur_idx[g] = j; cur_val[g] = fmaxf(v, 0.0f); ++g;
        } else if (key == thr) {
          if (e < need) { cur_idx[G + e] = j; cur_val[G + e] = fmaxf(v, 0.0f); }
          ++e;
        }
      }
    }
    __syncthreads();

    // 6) emit sparse z_t; carry state; final step scatters z_last
    if (tid < K_TOP) {
      size_t base = (size_t)(b * T_STEPS + t) * K_TOP;
      z_idx[base + tid] = cur_idx[tid];
      z_val[base + tid] = cur_val[tid];
      if (t == T_STEPS - 1)
        zlast[(size_t)b * D_SAE + cur_idx[tid]] = cur_val[tid];
      prev_idx[tid] = cur_idx[tid];
      prev_val[tid] = cur_val[tid];
    }
    __syncthreads();
  }
}

// ---------------------------------------------------------------------------
// Sparse decode: x_hat[bt,:] = b_dec + sum_j z_val[j] * W_dec[z_idx[j], :]
// plus per-(b,t) squared-error partial sum (deterministic tree reduce).
// ---------------------------------------------------------------------------
__global__ __launch_bounds__(256)
void decode_kernel(const int* __restrict__ z_idx, const float* __restrict__ z_val,
                   const float* __restrict__ W_dec, const float* __restrict__ b_dec,
                   const float* __restrict__ x, float* __restrict__ xhat,
                   float* __restrict__ partial) {
  __shared__ int   sidx[K_TOP];
  __shared__ float sval[K_TOP];
  __shared__ float red[256];
  const int bt = blockIdx.x, tid = threadIdx.x;
  if (tid < K_TOP) {
    sidx[tid] = z_idx[(size_t)bt * K_TOP + tid];
    sval[tid] = z_val[(size_t)bt * K_TOP + tid];
  }
  __syncthreads();

  float a0 = b_dec[tid], a1 = b_dec[tid + 256], a2 = b_dec[tid + 512];
  for (int j = 0; j < K_TOP; ++j) {
    float v = sval[j];
    if (v != 0.0f) {
      const float* wr = W_dec + (size_t)sidx[j] * D_INP;
      a0 += v * wr[tid];
      a1 += v * wr[tid + 256];
      a2 += v * wr[tid + 512];
    }
  }
  const size_t base = (size_t)bt * D_INP;
  xhat[base + tid]       = a0;
  xhat[base + tid + 256] = a1;
  xhat[base + tid + 512] = a2;

  float d0 = a0 - x[base + tid];
  float d1 = a1 - x[base + tid + 256];
  float d2 = a2 - x[base + tid + 512];
  red[tid] = d0 * d0 + d1 * d1 + d2 * d2;
  __syncthreads();
  for (int off = 128; off > 0; off >>= 1) {
    if (tid < off) red[tid] += red[tid + off];
    __syncthreads();
  }
  if (tid == 0) partial[bt] = red[0];
}

__global__ __launch_bounds__(256)
void loss_reduce(const float* __restrict__ partial, float* __restrict__ out) {
  __shared__ float red[256];
  const int tid = threadIdx.x;
  float s = 0.0f;
  for (int i = tid; i < M_TOTAL; i += 256) s += partial[i];   // fixed order
  red[tid] = s; __syncthreads();
  for (int off = 128; off > 0; off >>= 1) {
    if (tid < off) red[tid] += red[tid + off];
    __syncthreads();
  }
  if (tid == 0) out[0] = red[0] / (float)M_TOTAL;
}

__global__ __launch_bounds__(256)
void zero_kernel(float* __restrict__ p, int n) {
  int i = blockIdx.x * 256 + threadIdx.x;
  if (i < n) p[i] = 0.0f;
}

// ---------------------------------------------------------------------------
extern "C" void kernel_launch(void* const* d_in, const int* in_sizes, int n_in,
                              void* d_out, int out_size, void* d_ws, size_t ws_size,
                              hipStream_t stream) {
  const float* x        = (const float*)d_in[0];
  const float* W_enc    = (const float*)d_in[1];
  const float* W_dec    = (const float*)d_in[2];
  const float* b_enc    = (const float*)d_in[3];
  const float* b_dec    = (const float*)d_in[4];
  const float* gate_raw = (const float*)d_in[5];

  float* out   = (float*)d_out;
  float* loss  = out;
  float* xhat  = out + 1;
  float* zlast = out + 1 + (size_t)M_TOTAL * D_INP;

  char*  ws  = (char*)d_ws;
  size_t off = 0;
  float* pre = (float*)(ws + off); off += (size_t)M_TOTAL * D_SAE * sizeof(float);
  int*   zi  = (int*)  (ws + off); off += (size_t)M_TOTAL * K_TOP * sizeof(int);
  float* zv  = (float*)(ws + off); off += (size_t)M_TOTAL * K_TOP * sizeof(float);
  float* prt = (float*)(ws + off);

  encode_wmma<<<dim3(D_SAE / 256, M_TOTAL / 32), 128, 0, stream>>>(x, W_enc, b_enc, pre);
  zero_kernel<<<(BATCH * D_SAE + 255) / 256, 256, 0, stream>>>(zlast, BATCH * D_SAE);
  scan_topk<<<BATCH, 512, 0, stream>>>(pre, gate_raw, zi, zv, zlast);
  decode_kernel<<<M_TOTAL, 256, 0, stream>>>(zi, zv, W_dec, b_dec, x, xhat, prt);
  loss_reduce<<<1, 256, 0, stream>>>(prt, loss);
}